// WorkingMemory_17154099380755
// MI455X (gfx1250) — compile-verified
//
#include <hip/hip_runtime.h>
#include <hip/hip_bf16.h>

typedef __attribute__((ext_vector_type(2))) float v2f;
typedef __attribute__((ext_vector_type(8))) float v8f;

constexpr int NB = 64;    // batch
constexpr int NM = 1024;  // memory slots
constexpr int NW = 64;    // word width
constexpr int NR = 4;     // read heads
constexpr int XI = 470;   // xi dim
#define EPSF 1e-6f

// ---- workspace layout (float offsets) ----
constexpr size_t WS_WW  = 0;                           // NB*NM
constexpr size_t WS_A   = WS_WW  + (size_t)NB*NM;      // NB*NM*NR
constexpr size_t WS_B   = WS_A   + (size_t)NB*NM*NR;   // NB*NM*NR
constexpr size_t WS_ATP = WS_B   + (size_t)NB*NM*NR;   // NB*4*NM*NR (4 row-chunk partials)
constexpr size_t WS_BTP = WS_ATP + (size_t)NB*4*NM*NR; // NB*4*NM*NR
constexpr size_t WS_C   = WS_BTP + (size_t)NB*4*NM*NR; // NB*NM*NR
constexpr size_t WS_S   = WS_C   + (size_t)NB*NM*NR;   // NB*NR
constexpr size_t WS_P   = WS_S   + (size_t)NB*NR;      // NB*NR
constexpr size_t WS_ER  = WS_P   + (size_t)NB*NR;      // NB*NW
constexpr size_t WS_WV  = WS_ER  + (size_t)NB*NW;      // NB*NW
constexpr size_t WS_RM  = WS_WV  + (size_t)NB*NW;      // NB*12

__device__ __forceinline__ float sigmoidf_(float x) { return 1.0f / (1.0f + __expf(-x)); }
__device__ __forceinline__ float softplusf_(float x) { return (x > 20.f) ? x : log1pf(__expf(x)); }

// ================= Kernel 1: per-batch setup =================
__global__ __launch_bounds__(1024)
void k1_setup(const float* __restrict__ xi, const float* __restrict__ mem,
              const float* __restrict__ twg, const float* __restrict__ rw,
              const float* __restrict__ usage, float* __restrict__ ws)
{
    const int b = blockIdx.x;
    const int j = threadIdx.x;                 // one thread per memory row
    __shared__ float wkeyL[NW], eraseL[NW], wvecL[NW], rkL[NW * NR];
    __shared__ float scal[40];
    __shared__ float skey[NM];
    __shared__ int   sidx[NM];
    __shared__ float sval[NM];

    const float* xb = xi + (size_t)b * XI;
    if (j < NW) {
        wkeyL[j]  = xb[260 + j];
        eraseL[j] = sigmoidf_(xb[325 + j]);
        wvecL[j]  = xb[389 + j];
    }
    if (j < NW * NR) rkL[j] = xb[j];           // rk[w][r] = xb[w*4+r]
    if (j == 0) {
        scal[0] = 1.0f + softplusf_(xb[324]);  // write strength (oneplus)
        scal[1] = sigmoidf_(xb[457]);          // allocation gate
        scal[2] = sigmoidf_(xb[458]);          // write gate
    }
    if (j < NR) scal[3 + j] = sigmoidf_(xb[453 + j]);   // free gates
    __syncthreads();
    if (j == 0) {
        float s = 0.f;
        for (int k = 0; k < NW; k++) s += wkeyL[k] * wkeyL[k];
        scal[19] = 1.0f / (sqrtf(s) + EPSF);
    }
    if (j < NR) {                               // read-key inverse norms
        float s = 0.f;
        for (int k = 0; k < NW; k++) { float v = rkL[k * NR + j]; s += v * v; }
        scal[20 + j] = 1.0f / (sqrtf(s) + EPSF);
    }
    if (j < NR) {                               // read modes: softmax over 3
        float a0 = xb[458 + 0 * NR + j], a1 = xb[458 + 1 * NR + j], a2 = xb[458 + 2 * NR + j];
        float mx = fmaxf(a0, fmaxf(a1, a2));
        float e0 = __expf(a0 - mx), e1 = __expf(a1 - mx), e2 = __expf(a2 - mx);
        float inv = 1.0f / (e0 + e1 + e2);
        scal[7 + 0 * NR + j] = e0 * inv;
        scal[7 + 1 * NR + j] = e1 * inv;
        scal[7 + 2 * NR + j] = e2 * inv;
    }
    __syncthreads();

    // phi = cosine(memory, write_key)
    const float* mrow = mem + ((size_t)b * NM + j) * NW;
    float nm = 0.f, dotk = 0.f;
    for (int k = 0; k < NW; k += 4) {
        float4 v = *(const float4*)(mrow + k);
        nm   += v.x * v.x + v.y * v.y + v.z * v.z + v.w * v.w;
        dotk += v.x * wkeyL[k] + v.y * wkeyL[k + 1] + v.z * wkeyL[k + 2] + v.w * wkeyL[k + 3];
    }
    float phi = dotk * scal[19] / (sqrtf(nm) + EPSF);

    float rwj[NR];
    for (int r = 0; r < NR; r++) rwj[r] = rw[((size_t)b * NM + j) * NR + r];
    float psi = 1.f;
    for (int r = 0; r < NR; r++) psi *= (1.f - scal[3 + r] * rwj[r]);
    float u  = usage[(size_t)b * NM + j];
    float st = scal[0];
    float us = (u + st - u * st) * psi;

    // ---- bitonic sort ascending (key=usage, payload=index) ----
    skey[j] = us; sidx[j] = j;
    __syncthreads();
    for (int k = 2; k <= NM; k <<= 1)
        for (int s = k >> 1; s > 0; s >>= 1) {
            int ixj = j ^ s;
            if (ixj > j) {
                bool up = ((j & k) == 0);
                float a = skey[j], c = skey[ixj];
                if ((a > c) == up) {
                    skey[j] = c; skey[ixj] = a;
                    int t = sidx[j]; sidx[j] = sidx[ixj]; sidx[ixj] = t;
                }
            }
            __syncthreads();
        }
    float su = skey[j];
    // inclusive multiplicative scan (cumprod of sorted usage)
    for (int off = 1; off < NM; off <<= 1) {
        float t = (j >= off) ? skey[j - off] : 1.0f;
        __syncthreads();
        skey[j] *= t;
        __syncthreads();
    }
    float allocSorted = (1.f - su) * skey[j];
    sval[sidx[j]] = allocSorted;               // scatter by inverse permutation
    __syncthreads();
    float alloc = sval[j];
    float ag = scal[1], wg = scal[2];
    float wwj = wg * (ag * alloc + (1.f - ag) * phi);
    ws[WS_WW + (size_t)b * NM + j] = wwj;

    // ---- S[r] = sum tw*rw, P[r] = sum ww*rw ----
    float twj = twg[(size_t)b * NM + j];
    for (int r = 0; r < NR; r++) {
        __syncthreads();
        sval[j] = twj * rwj[r];
        __syncthreads();
        for (int s = 512; s > 0; s >>= 1) { if (j < s) sval[j] += sval[j + s]; __syncthreads(); }
        if (j == 0) ws[WS_S + (size_t)b * NR + r] = sval[0];
        __syncthreads();
        sval[j] = wwj * rwj[r];
        __syncthreads();
        for (int s = 512; s > 0; s >>= 1) { if (j < s) sval[j] += sval[j + s]; __syncthreads(); }
        if (j == 0) ws[WS_P + (size_t)b * NR + r] = sval[0];
    }

    // ---- content weights c = softmax_m(cos(memory_new, rk) * rw) ----
    float nmn = 0.f, drk[NR] = {0.f, 0.f, 0.f, 0.f};
    for (int k = 0; k < NW; k += 4) {
        float4 v = *(const float4*)(mrow + k);
        float mv[4] = {v.x, v.y, v.z, v.w};
#pragma unroll
        for (int u4 = 0; u4 < 4; u4++) {
            int c0 = k + u4;
            float mn2 = mv[u4] * (1.f - wwj * eraseL[c0]) + wwj * wvecL[c0];
            nmn += mn2 * mn2;
#pragma unroll
            for (int r = 0; r < NR; r++) drk[r] += mn2 * rkL[c0 * NR + r];
        }
    }
    float invn = 1.0f / (sqrtf(nmn) + EPSF);
    float score[NR];
    for (int r = 0; r < NR; r++) score[r] = drk[r] * invn * scal[20 + r] * rwj[r];
    for (int r = 0; r < NR; r++) {
        __syncthreads();
        sval[j] = score[r];
        __syncthreads();
        for (int s = 512; s > 0; s >>= 1) { if (j < s) sval[j] = fmaxf(sval[j], sval[j + s]); __syncthreads(); }
        float mx = sval[0];
        __syncthreads();
        float e = __expf(score[r] - mx);
        sval[j] = e;
        __syncthreads();
        for (int s = 512; s > 0; s >>= 1) { if (j < s) sval[j] += sval[j + s]; __syncthreads(); }
        float sm = sval[0];
        ws[WS_C + ((size_t)b * NM + j) * NR + r] = e / sm;
    }

    if (j < NW) { ws[WS_ER + (size_t)b * NW + j] = eraseL[j]; ws[WS_WV + (size_t)b * NW + j] = wvecL[j]; }
    if (j < 12) ws[WS_RM + (size_t)b * 12 + j] = scal[7 + j];
}

// ========== Kernel 2 (fused, single pass over T) ==========
// Block: batch b, row-chunk rc (256 rows). Loops 8 col-chunks of 128.
// Tile staged in LDS via CDNA5 async global->LDS copies (ASYNCcnt).
// Waves 4-7: row side -> A,B (full, direct store).
// Waves 0-3: col side -> At,Bt partial per row-chunk (direct store, summed in k3).
constexpr int TROWS = 256;
constexpr int TCOLS = 128;
constexpr int TSTR  = 132;   // padded LDS row stride in floats (528B, 16B aligned)

__global__ __launch_bounds__(256)
void k2_fused(const float* __restrict__ T, const float* __restrict__ rw,
              float* __restrict__ ws)
{
    extern __shared__ float smem[];
    float* tile = smem;                       // TROWS*TSTR floats
    float* rwL  = smem + TROWS * TSTR;        // NM*NR
    float* wwL  = rwL + NM * NR;              // NM

    const int b  = blockIdx.x >> 2;
    const int rc = blockIdx.x & 3;
    const int r0 = rc * TROWS;
    const int t  = threadIdx.x;
    const int wave = t >> 5, lane = t & 31;

    for (int i = t; i < NM; i += 256) {
        wwL[i] = ws[WS_WW + (size_t)b * NM + i];
        *(float4*)(rwL + i * NR) = *(const float4*)(rw + ((size_t)b * NM + i) * NR);
    }
    __syncthreads();

    const float* Tb = T + (size_t)b * NM * NM;
    const uint32_t tileBase = (uint32_t)(uintptr_t)tile;

    // row-side accumulators: lane owns rows lr1, lr1+32 for waves 4..7
    float accA[2][NR], accB[2][NR];
#pragma unroll
    for (int h = 0; h < 2; h++)
#pragma unroll
        for (int r = 0; r < NR; r++) { accA[h][r] = 0.f; accB[h][r] = 0.f; }
    const int lr1 = (wave - 4) * 64 + lane;   // valid when wave>=4

    for (int cc = 0; cc < 8; cc++) {
        // ---- stage 256x128 tile via async global->LDS b128 copies ----
        // 8192 16B segments; 32 per thread; consecutive lanes -> consecutive 16B (coalesced)
#pragma unroll 4
        for (int i = 0; i < 32; i++) {
            const int seg = i * 256 + t;
            const int row = seg >> 5;          // 32 segments (512B) per row
            const int s16 = seg & 31;
            const unsigned long long ga = (unsigned long long)(uintptr_t)
                (Tb + (size_t)(r0 + row) * NM + cc * TCOLS + s16 * 4);
            const uint32_t lo = tileBase + (uint32_t)(row * TSTR * 4 + s16 * 16);
            asm volatile("global_load_async_to_lds_b128 %0, %1, off"
                         :: "v"(lo), "v"(ga) : "memory");
        }
        // prefetch next chunk into L2 while waiting
        __builtin_prefetch(Tb + (size_t)(r0 + t) * NM + ((cc < 7) ? (cc + 1) : cc) * TCOLS, 0, 1);
        asm volatile("s_wait_asynccnt 0" ::: "memory");
        __syncthreads();

        if (wave < 4) {
            // ---- col side: thread t<128 owns column t of this chunk ----
            if (t < TCOLS) {
                float at[NR] = {0.f, 0.f, 0.f, 0.f}, bt[NR] = {0.f, 0.f, 0.f, 0.f};
                for (int lr = 0; lr < TROWS; lr++) {
                    const float val = tile[lr * TSTR + t];
                    const int rowg = r0 + lr;
                    const float* rp = rwL + rowg * NR;
                    const float vw = val * wwL[rowg];
#pragma unroll
                    for (int r = 0; r < NR; r++) { at[r] += val * rp[r]; bt[r] += vw * rp[r]; }
                }
                const int colg = cc * TCOLS + t;
                const size_t o = (((size_t)b * 4 + rc) * NM + colg) * NR;
#pragma unroll
                for (int r = 0; r < NR; r++) { ws[WS_ATP + o + r] = at[r]; ws[WS_BTP + o + r] = bt[r]; }
            }
        } else {
            // ---- row side: lane owns rows lr1 and lr1+32 across all chunks ----
#pragma unroll 4
            for (int c4 = 0; c4 < 32; c4++) {
                float4 v1 = *(const float4*)(tile + lr1 * TSTR + c4 * 4);
                float4 v2 = *(const float4*)(tile + (lr1 + 32) * TSTR + c4 * 4);
                float tv1[4] = {v1.x, v1.y, v1.z, v1.w};
                float tv2[4] = {v2.x, v2.y, v2.z, v2.w};
                const int colg = cc * TCOLS + c4 * 4;
#pragma unroll
                for (int u = 0; u < 4; u++) {
                    const float* rp = rwL + (colg + u) * NR;
                    const float wv = wwL[colg + u];
                    const float a1 = tv1[u], a2 = tv2[u];
                    const float w1 = a1 * wv, w2 = a2 * wv;
#pragma unroll
                    for (int r = 0; r < NR; r++) {
                        accA[0][r] += a1 * rp[r]; accB[0][r] += w1 * rp[r];
                        accA[1][r] += a2 * rp[r]; accB[1][r] += w2 * rp[r];
                    }
                }
            }
        }
        __syncthreads();
    }

    if (wave >= 4) {
#pragma unroll
        for (int h = 0; h < 2; h++) {
            const int rowg = r0 + lr1 + h * 32;
            const size_t o = ((size_t)b * NM + rowg) * NR;
#pragma unroll
            for (int r = 0; r < NR; r++) { ws[WS_A + o + r] = accA[h][r]; ws[WS_B + o + r] = accB[h][r]; }
        }
    }
}

// ========== Kernel 3: rw_new assembly + WMMA contraction ==========
// out[b][w][r] = sum_m memory_new[b][m][w] * rw_new[b][m][r]
__global__ __launch_bounds__(256)
void k3_final(const float* __restrict__ T, const float* __restrict__ mem,
              const float* __restrict__ twg, const float* __restrict__ rw,
              const float* __restrict__ ws, float* __restrict__ out)
{
    const int b = blockIdx.x;
    const int t = threadIdx.x;
    __shared__ float rwnL[NM * NR];        // rw_new, 16 KB
    __shared__ float mnL[64 * 68];         // memory_new chunk (64 rows x 64 cols, stride 68)
    __shared__ float eraseL[NW], wvecL[NW], rmL[12], SL[NR], PL[NR];
    __shared__ float outL[8 * 16 * 4];

    if (t < NW) { eraseL[t] = ws[WS_ER + (size_t)b * NW + t]; wvecL[t] = ws[WS_WV + (size_t)b * NW + t]; }
    if (t < 12) rmL[t] = ws[WS_RM + (size_t)b * 12 + t];
    if (t < NR) { SL[t] = ws[WS_S + (size_t)b * NR + t]; PL[t] = ws[WS_P + (size_t)b * NR + t]; }
    __syncthreads();

    // Phase 1: fwd/bwd/rw_new per row (each thread does 4 rows)
    for (int q = 0; q < 4; q++) {
        const int j = t + q * 256;
        const size_t jr = ((size_t)b * NM + j) * NR;
        const float wwj = ws[WS_WW + (size_t)b * NM + j];
        const float twj = twg[(size_t)b * NM + j];
        const float Tjj = T[(size_t)b * NM * NM + (size_t)j * NM + j];
#pragma unroll
        for (int r = 0; r < NR; r++) {
            const float rwv  = rw[jr + r];
            const float diag = ((1.f - 2.f * wwj) * Tjj + wwj * twj) * rwv;
            float at = 0.f, bt = 0.f;
#pragma unroll
            for (int p = 0; p < 4; p++) {
                const size_t po = (((size_t)b * 4 + p) * NM + j) * NR + r;
                at += ws[WS_ATP + po];
                bt += ws[WS_BTP + po];
            }
            const float fwd = (1.f - wwj) * ws[WS_A + jr + r] - ws[WS_B + jr + r] + wwj * SL[r] - diag;
            const float bwd = (1.f - wwj) * at - bt + twj * PL[r] - diag;
            const float cc  = ws[WS_C + jr + r];
            rwnL[j * NR + r] = rmL[0 * NR + r] * bwd + rmL[1 * NR + r] * cc + rmL[2 * NR + r] * fwd;
        }
    }
    __syncthreads();

    // Phase 2: WMMA f32 16x16x4. 8 waves: wtile = wave&3 (16 w-rows), half = wave>>2 (K split)
    const int wave = t >> 5, lane = t & 31;
    const int wtile = wave & 3, half = wave >> 2;
    const int l2 = lane & 15, hi = lane >> 4;
    v8f acc;
#pragma unroll
    for (int i = 0; i < 8; i++) acc[i] = 0.0f;

    for (int chunk = 0; chunk < 16; chunk++) {
        // stage 64 rows of memory_new into LDS (transform applied on the fly)
        const int lr = t >> 2;               // local row 0..63
        const int cb = (t & 3) * 16;         // col base
        const int gk = chunk * 64 + lr;
        const float wwk = ws[WS_WW + (size_t)b * NM + gk];
        const float* mr = mem + ((size_t)b * NM + gk) * NW + cb;
#pragma unroll
        for (int i = 0; i < 16; i += 4) {
            float4 v = *(const float4*)(mr + i);
            float mv[4] = {v.x, v.y, v.z, v.w};
#pragma unroll
            for (int u = 0; u < 4; u++) {
                const int c0 = cb + i + u;
                mnL[lr * 68 + c0] = mv[u] * (1.f - wwk * eraseL[c0]) + wwk * wvecL[c0];
            }
        }
        __syncthreads();
#pragma unroll
        for (int s = 0; s < 8; s++) {
            const int kl = half * 32 + s * 4;        // local K base within chunk
            const int ka = kl + (hi ? 2 : 0);
            v2f afrag;
            afrag.x = mnL[(ka + 0) * 68 + wtile * 16 + l2];   // A[M=l2][K=ka]
            afrag.y = mnL[(ka + 1) * 68 + wtile * 16 + l2];   // A[M=l2][K=ka+1]
            const int kg = chunk * 64 + ka;
            const float b0v = rwnL[(kg + 0) * NR + (l2 & 3)];
            const float b1v = rwnL[(kg + 1) * NR + (l2 & 3)];
            v2f bfrag;
            bfrag.x = (l2 < NR) ? b0v : 0.0f;                 // B[K=kg][N=l2]
            bfrag.y = (l2 < NR) ? b1v : 0.0f;                 // B[K=kg+1][N=l2]
            acc = __builtin_amdgcn_wmma_f32_16x16x4_f32(
                false, afrag, false, bfrag, (short)0, acc, false, false);
        }
        __syncthreads();
    }

    // Phase 3: combine the two K-halves per wtile and write out
#pragma unroll
    for (int v = 0; v < 8; v++) {
        const int mrow = v + hi * 8;         // D: VGPR v, lanes>=16 hold M=v+8
        if (l2 < NR) outL[wave * 64 + mrow * 4 + l2] = acc[v];
    }
    __syncthreads();
    {
        const int wt = t >> 6;
        const int row = (t >> 2) & 15;
        const int r = t & 3;
        const float v = outL[wt * 64 + row * 4 + r] + outL[(wt + 4) * 64 + row * 4 + r];
        out[((size_t)b * NW + (wt * 16 + row)) * NR + r] = v;
    }
}

extern "C" void kernel_launch(void* const* d_in, const int* in_sizes, int n_in,
                              void* d_out, int out_size, void* d_ws, size_t ws_size,
                              hipStream_t stream) {
    const float* xi  = (const float*)d_in[0];
    const float* mem = (const float*)d_in[1];
    const float* T   = (const float*)d_in[2];
    const float* tw  = (const float*)d_in[3];
    const float* rw  = (const float*)d_in[4];
    const float* ur  = (const float*)d_in[5];
    float* ws  = (float*)d_ws;
    float* out = (float*)d_out;

    const size_t k2Lds = (size_t)(TROWS * TSTR + NM * NR + NM) * sizeof(float); // ~152 KB
    static bool attrSet = false;
    (void)hipFuncSetAttribute(reinterpret_cast<const void*>(k2_fused),
                              hipFuncAttributeMaxDynamicSharedMemorySize, (int)k2Lds);
    (void)attrSet;

    k1_setup<<<NB, 1024, 0, stream>>>(xi, mem, tw, rw, ur, ws);
    k2_fused<<<NB * 4, 256, k2Lds, stream>>>(T, rw, ws);
    k3_final<<<NB, 256, 0, stream>>>(T, mem, tw, rw, ws, out);
}